// StandardGCN_26551487824427
// MI455X (gfx1250) — compile-verified
//
#include <hip/hip_runtime.h>

// ---------------------------------------------------------------------------
// StandardGCN on MI455X (gfx1250, wave32)
//   3 x [ GEMM (WMMA f32 16x16x4)  ->  edge gather/scatter-add  -> BN+ReLU ]
// Edge scatter (~1.8 GB irregular traffic) is the roofline term; GEMMs use
// the exact-precision f32 WMMA path, fully unrolled and branch-free (W3 is
// zero-padded to a multiple of 16 columns so no lane guards are needed).
// K and D are compile-time so all strided accesses use immediate offsets.
// ---------------------------------------------------------------------------

typedef __attribute__((ext_vector_type(2))) float v2f;
typedef __attribute__((ext_vector_type(8))) float v8f;

#define GCN_N    100000
#define GCN_E    1200000
#define GCN_IN   128
#define GCN_H    64
#define GCN_OUT  40
#define GCN_OUTP 48   // OUT padded to multiple of 16

// ---------------------------------------------------------------------------
// GEMM: Out[N x D] = A[N x K] @ W[K x D], one wave per 16x16 tile.
// blockIdx.x = row tile (N/16 tiles, N % 16 == 0), wave-in-block = col tile.
// D % 16 == 0 (W padded beforehand). No divergence anywhere: EXEC is all-1s
// at every WMMA as the ISA requires.
// V_WMMA_F32_16X16X4_F32 VGPR layouts (ISA 7.12.2, 32-bit):
//   A 16x4 : lane L<16 -> {K0,K1} of row L ; lane L>=16 -> {K2,K3} of row L-16
//   B 4x16 : lane L<16 -> col L of rows {K0,K1} ; lanes 16-31 -> rows {K2,K3}
//   C/D    : VGPR v, lane L -> row m0 + v + 8*(L/16), col n0 + (L%16)
// ---------------------------------------------------------------------------
template <int K, int D>
__global__ void gcn_gemm_wmma(const float* __restrict__ A,
                              const float* __restrict__ W,
                              float* __restrict__ Out) {
  const int lane = threadIdx.x & 31;
  const int wv   = threadIdx.x >> 5;           // col tile = wave in block
  const int m0   = blockIdx.x * 16;
  const int n0   = wv * 16;
  const int r    = lane & 15;
  const int kh   = lane >> 4;                  // 0: K pair {0,1}, 1: {2,3}

  const float* arow = A + (long long)(m0 + r) * K + 2 * kh;
  const float* wcol = W + 2 * kh * D + (n0 + r);

  v8f acc = {};
#pragma unroll
  for (int k = 0; k < K; k += 4) {
    v2f a = *reinterpret_cast<const v2f*>(arow + k);     // 8B aligned
    v2f b;
    b.x = wcol[(k + 0) * D];                             // immediate offsets
    b.y = wcol[(k + 1) * D];
    // 8 args: (neg_a, A, neg_b, B, c_mod, C, reuse_a, reuse_b)
    acc = __builtin_amdgcn_wmma_f32_16x16x4_f32(
        false, a, false, b, (short)0, acc, false, false);
  }

  float* orow = Out + (long long)(m0 + 8 * kh) * D + (n0 + r);
#pragma unroll
  for (int v = 0; v < 8; ++v)
    orow[v * D] = acc[v];
}

// ---------------------------------------------------------------------------
// Zero-pad W3 (64 x 40) -> (64 x 48); padded cols are exact zeros.
// ---------------------------------------------------------------------------
__global__ void gcn_pad_w3(const float* __restrict__ W3,
                           float* __restrict__ Wp) {
  int i = blockIdx.x * blockDim.x + threadIdx.x;
  if (i >= GCN_H * GCN_OUTP) return;
  int k = i / GCN_OUTP;
  int c = i - k * GCN_OUTP;
  Wp[i] = (c < GCN_OUT) ? W3[k * GCN_OUT + c] : 0.0f;
}

// ---------------------------------------------------------------------------
// Edge scatter: agg[dst[e]][0:D] += h[src[e]][0:D] * ew[e]
// One thread per (edge, 4-float chunk); float4 gather, hw f32 atomics.
// ldH = row stride of h (may exceed D for the padded layer-3 buffer).
// ---------------------------------------------------------------------------
__global__ void gcn_edge_scatter(const float* __restrict__ h,
                                 const int* __restrict__ src,
                                 const int* __restrict__ dst,
                                 const float* __restrict__ ew,
                                 float* __restrict__ agg,
                                 long long nEdges, int ldH, int D) {
  const int chunks = D >> 2;
  long long tid = (long long)blockIdx.x * blockDim.x + threadIdx.x;
  long long total = nEdges * chunks;
  if (tid >= total) return;

  long long e = tid / chunks;
  int c = (int)(tid - e * chunks) << 2;

  const int s = src[e];
  const int d = dst[e];
  const float w = ew[e];

  const float4 m = *reinterpret_cast<const float4*>(h + (long long)s * ldH + c);
  float* ap = agg + (long long)d * D + c;
  unsafeAtomicAdd(ap + 0, m.x * w);
  unsafeAtomicAdd(ap + 1, m.y * w);
  unsafeAtomicAdd(ap + 2, m.z * w);
  unsafeAtomicAdd(ap + 3, m.w * w);
}

// ---------------------------------------------------------------------------
// BN stats: stats[0..63] = column sums, stats[64..127] = column sum-of-squares.
// 256 threads = 4 row-lanes x 64 columns; LDS partial reduce then f32 atomics.
// ---------------------------------------------------------------------------
__global__ void gcn_bn_stats(const float* __restrict__ x,
                             float* __restrict__ stats, int N) {
  const int col  = threadIdx.x & 63;
  const int rsub = threadIdx.x >> 6;      // 0..3

  float s = 0.0f, ss = 0.0f;
  for (long long row = (long long)blockIdx.x * 4 + rsub; row < N;
       row += (long long)gridDim.x * 4) {
    float v = x[row * GCN_H + col];
    s  += v;
    ss += v * v;
  }

  __shared__ float shS[4][64];
  __shared__ float shQ[4][64];
  shS[rsub][col] = s;
  shQ[rsub][col] = ss;
  __syncthreads();

  if (rsub == 0) {
    s  = shS[0][col] + shS[1][col] + shS[2][col] + shS[3][col];
    ss = shQ[0][col] + shQ[1][col] + shQ[2][col] + shQ[3][col];
    unsafeAtomicAdd(&stats[col], s);
    unsafeAtomicAdd(&stats[64 + col], ss);
  }
}

// ---------------------------------------------------------------------------
// BN apply + ReLU.  The pre-BN conv bias cancels inside training-mode BN
// (x - mean is bias invariant), so it is intentionally never added.
// ---------------------------------------------------------------------------
__global__ void gcn_bn_relu(const float* __restrict__ x,
                            const float* __restrict__ stats,
                            const float* __restrict__ g,
                            const float* __restrict__ beta,
                            float* __restrict__ y,
                            long long total, float invN) {
  long long i = (long long)blockIdx.x * blockDim.x + threadIdx.x;
  if (i >= total) return;
  const int col = (int)(i & (GCN_H - 1));
  const float mu  = stats[col] * invN;
  const float var = stats[GCN_H + col] * invN - mu * mu;
  const float sc  = g[col] * rsqrtf(var + 1e-5f);
  const float v   = (x[i] - mu) * sc + beta[col];
  y[i] = fmaxf(v, 0.0f);
}

__global__ void gcn_zero(float* __restrict__ p, long long n) {
  long long i = (long long)blockIdx.x * blockDim.x + threadIdx.x;
  if (i < n) p[i] = 0.0f;
}

__global__ void gcn_init_bias(float* __restrict__ out,
                              const float* __restrict__ b,
                              long long total, int D) {
  long long i = (long long)blockIdx.x * blockDim.x + threadIdx.x;
  if (i < total) out[i] = b[i % D];
}

static inline int cdiv_ll(long long a, long long b) {
  return (int)((a + b - 1) / b);
}

extern "C" void kernel_launch(void* const* d_in, const int* in_sizes, int n_in,
                              void* d_out, int out_size, void* d_ws, size_t ws_size,
                              hipStream_t stream) {
  (void)in_sizes; (void)n_in; (void)out_size; (void)ws_size;
  const int N = GCN_N, E = GCN_E, H = GCN_H, OUT = GCN_OUT, OUTP = GCN_OUTP;

  const float* X   = (const float*)d_in[0];
  const int*   ei  = (const int*)d_in[1];
  const int*   src = ei;
  const int*   dst = ei + E;
  const float* ew  = (const float*)d_in[2];
  const float* W1  = (const float*)d_in[3];
  const float* g1  = (const float*)d_in[5];
  const float* be1 = (const float*)d_in[6];
  const float* W2  = (const float*)d_in[7];
  const float* g2  = (const float*)d_in[9];
  const float* be2 = (const float*)d_in[10];
  const float* W3  = (const float*)d_in[11];
  const float* b3  = (const float*)d_in[12];
  float* out = (float*)d_out;

  float* bufA  = (float*)d_ws;                       // N x 64
  float* bufB  = bufA + (long long)N * H;            // N x 64
  float* bufC  = bufB + (long long)N * H;            // N x 48 (padded h3)
  float* W3p   = bufC + (long long)N * OUTP;         // 64 x 48
  float* stats = W3p + H * OUTP;                     // 128

  const int rowTiles = N / 16;                       // 6250
  const long long NH = (long long)N * H;
  const long long NO = (long long)N * OUT;
  const float invN = 1.0f / (float)N;

  // ---- Layer 1: GCNConv(128->64) + BN + ReLU -----------------------------
  gcn_gemm_wmma<GCN_IN, GCN_H><<<rowTiles, 128, 0, stream>>>(X, W1, bufA);
  gcn_zero<<<cdiv_ll(NH, 256), 256, 0, stream>>>(bufB, NH);
  gcn_edge_scatter<<<cdiv_ll((long long)E * (H / 4), 256), 256, 0, stream>>>(
      bufA, src, dst, ew, bufB, E, H, H);
  gcn_zero<<<1, 128, 0, stream>>>(stats, 128);
  gcn_bn_stats<<<512, 256, 0, stream>>>(bufB, stats, N);
  gcn_bn_relu<<<cdiv_ll(NH, 256), 256, 0, stream>>>(bufB, stats, g1, be1,
                                                    bufA, NH, invN);

  // ---- Layer 2: GCNConv(64->64) + BN + ReLU ------------------------------
  gcn_gemm_wmma<GCN_H, GCN_H><<<rowTiles, 128, 0, stream>>>(bufA, W2, bufB);
  gcn_zero<<<cdiv_ll(NH, 256), 256, 0, stream>>>(bufA, NH);
  gcn_edge_scatter<<<cdiv_ll((long long)E * (H / 4), 256), 256, 0, stream>>>(
      bufB, src, dst, ew, bufA, E, H, H);
  gcn_zero<<<1, 128, 0, stream>>>(stats, 128);
  gcn_bn_stats<<<512, 256, 0, stream>>>(bufA, stats, N);
  gcn_bn_relu<<<cdiv_ll(NH, 256), 256, 0, stream>>>(bufA, stats, g2, be2,
                                                    bufB, NH, invN);

  // ---- Layer 3: GCNConv(64->40) + bias -----------------------------------
  gcn_pad_w3<<<cdiv_ll(H * OUTP, 256), 256, 0, stream>>>(W3, W3p);
  gcn_gemm_wmma<GCN_H, GCN_OUTP><<<rowTiles, 96, 0, stream>>>(bufB, W3p, bufC);
  gcn_init_bias<<<cdiv_ll(NO, 256), 256, 0, stream>>>(out, b3, NO, OUT);
  gcn_edge_scatter<<<cdiv_ll((long long)E * (OUT / 4), 256), 256, 0, stream>>>(
      bufC, src, dst, ew, out, E, OUTP, OUT);
}